// EdgeRelativeEmbed_22651657519354
// MI455X (gfx1250) — compile-verified
//
#include <hip/hip_runtime.h>
#include <hip/hip_bf16.h>

typedef float v2f __attribute__((ext_vector_type(2)));
typedef float v8f __attribute__((ext_vector_type(8)));

#define ADIM 128
#define VDIM 64

template <bool GUARD>
__global__ __launch_bounds__(256) void EdgeRelativeEmbed_kernel(
    const float* __restrict__ pos0, const float* __restrict__ pos1,
    const int*   __restrict__ src,  const int*   __restrict__ dst,
    const float* __restrict__ Wa,   const float* __restrict__ ba,
    const float* __restrict__ Wv,
    float* __restrict__ aout, float* __restrict__ vout,
    int B, int N, int E)
{
    const int lane = threadIdx.x & 31;
    const int n16  = lane & 15;
    const bool hi  = lane >= 16;   // lanes 16..31 carry K=2,3 of each K-step

    // ---- Per-wave constant fragments (reused across all tiles) ----
    // B-fragment layout for V_WMMA_F32_16X16X4_F32 (4x16, K x N):
    //   VGPR0: lanes0-15 -> K=0, lanes16-31 -> K=2 ; VGPR1: K=1 / K=3
    // K=6 padded to 8 => two K-steps; step1 lanes>=16 cover k=6,7 => zero.
    v2f Bwa[8][2];
#pragma unroll
    for (int nt = 0; nt < 8; ++nt) {
        int n  = nt * 16 + n16;
        int k0 = hi ? 2 : 0;
        v2f f0; f0.x = Wa[n * 6 + k0]; f0.y = Wa[n * 6 + k0 + 1];
        Bwa[nt][0] = f0;
        v2f f1; f1.x = hi ? 0.0f : Wa[n * 6 + 4]; f1.y = hi ? 0.0f : Wa[n * 6 + 5];
        Bwa[nt][1] = f1;
    }
    v2f Bwv[4][2];
    const float third = 1.0f / 3.0f;   // fold /3 into W_v
#pragma unroll
    for (int nt = 0; nt < 4; ++nt) {
        int n  = nt * 16 + n16;
        int k0 = hi ? 2 : 0;
        v2f f0; f0.x = third * Wv[n * 6 + k0]; f0.y = third * Wv[n * 6 + k0 + 1];
        Bwv[nt][0] = f0;
        v2f f1; f1.x = hi ? 0.0f : third * Wv[n * 6 + 4]; f1.y = hi ? 0.0f : third * Wv[n * 6 + 5];
        Bwv[nt][1] = f1;
    }
    float bias[8];
#pragma unroll
    for (int nt = 0; nt < 8; ++nt) bias[nt] = ba[nt * 16 + n16];

    // ---- Grid-stride loop over 16-edge tiles (32-bit indices) ----
    const int tilesPerBatch = (E + 15) >> 4;
    const int totalTiles = B * tilesPerBatch;
    const int wavesPerBlock = blockDim.x >> 5;
    int wave = blockIdx.x * wavesPerBlock + (threadIdx.x >> 5);
    const int waveStride = gridDim.x * wavesPerBlock;
    const int hi8 = hi ? 8 : 0;

    for (int t = wave; t < totalTiles; t += waveStride) {
        const int b  = t / tilesPerBatch;
        const int e0 = (t - b * tilesPerBatch) << 4;
        int e = e0 + n16;
        if (GUARD) { if (e >= E) e = E - 1; }  // clamp (keeps EXEC all-1 for WMMA)
        const int si = src[e];
        const int di = dst[e];

        const float* p0 = pos0 + (size_t)b * N * 3;
        const float* p1 = pos1 + (size_t)b * N * 3;
        float p0s[3], p0d[3], p1s[3], p1d[3];
#pragma unroll
        for (int k = 0; k < 3; ++k) {
            p0s[k] = p0[(size_t)si * 3 + k];
            p0d[k] = p0[(size_t)di * 3 + k];
            p1s[k] = p1[(size_t)si * 3 + k];
            p1d[k] = p1[(size_t)di * 3 + k];
        }

        // 6 relative vectors in reference stack order
        float v[6][3];
#pragma unroll
        for (int k = 0; k < 3; ++k) {
            v[0][k] = p0d[k] - p0s[k];
            v[1][k] = p1d[k] - p1s[k];
            v[2][k] = p1s[k] - p0s[k];
            v[3][k] = p1d[k] - p0d[k];
            v[4][k] = p1s[k] - p0d[k];
            v[5][k] = p1d[k] - p0s[k];
        }
        float nr[6];
#pragma unroll
        for (int i = 0; i < 6; ++i)
            nr[i] = __builtin_amdgcn_sqrtf(
                v[i][0] * v[i][0] + v[i][1] * v[i][1] + v[i][2] * v[i][2]);

        // A-fragments (16x4 f32): VGPR0 = K0 (lo) / K2 (hi), VGPR1 = K1 / K3
        v2f An0; An0.x = hi ? nr[2] : nr[0]; An0.y = hi ? nr[3] : nr[1];
        v2f An1; An1.x = hi ? 0.0f : nr[4]; An1.y = hi ? 0.0f : nr[5];

        // Per-tile base pointers: all store offsets become compile-time
        // immediates (a_out: r*512B + nt*64B ; v_out: r*768B + nt*192B + k*4B)
        const size_t rowBase = (size_t)b * E + e0;
        float* __restrict__ abase = aout + (rowBase + hi8) * ADIM + n16;
        float* __restrict__ vbase = vout + ((rowBase + hi8) * VDIM + n16) * 3;

        // ---- a_out = norms @ W_a^T + b_a  (8 N-tiles x 2 K-steps) ----
#pragma unroll
        for (int nt = 0; nt < 8; ++nt) {
            v8f c;
#pragma unroll
            for (int i = 0; i < 8; ++i) c[i] = bias[nt];   // C[m][n] = b_a[n]
            c = __builtin_amdgcn_wmma_f32_16x16x4_f32(false, An0, false, Bwa[nt][0],
                                                      (short)0, c, false, false);
            c = __builtin_amdgcn_wmma_f32_16x16x4_f32(false, An1, false, Bwa[nt][1],
                                                      (short)0, c, false, false);
#pragma unroll
            for (int r = 0; r < 8; ++r) {
                if (GUARD && (e0 + r + hi8 >= E)) continue;
                abase[r * ADIM + nt * 16] = c[r];
            }
        }

        // ---- v_out = (W_v @ vecs)/3 : 3 spatial comps per N-tile so each
        //      lane stores 3 consecutive floats (dense 192B/half-wave) ----
#pragma unroll
        for (int nt = 0; nt < 4; ++nt) {
            v8f dk[3];
#pragma unroll
            for (int k = 0; k < 3; ++k) {
                v2f A0; A0.x = hi ? v[2][k] : v[0][k]; A0.y = hi ? v[3][k] : v[1][k];
                v2f A1; A1.x = hi ? 0.0f : v[4][k]; A1.y = hi ? 0.0f : v[5][k];
                v8f c = {0.f, 0.f, 0.f, 0.f, 0.f, 0.f, 0.f, 0.f};
                c = __builtin_amdgcn_wmma_f32_16x16x4_f32(false, A0, false, Bwv[nt][0],
                                                          (short)0, c, false, false);
                c = __builtin_amdgcn_wmma_f32_16x16x4_f32(false, A1, false, Bwv[nt][1],
                                                          (short)0, c, false, false);
                dk[k] = c;
            }
#pragma unroll
            for (int r = 0; r < 8; ++r) {
                if (GUARD && (e0 + r + hi8 >= E)) continue;
                vbase[r * VDIM * 3 + nt * 48 + 0] = dk[0][r];
                vbase[r * VDIM * 3 + nt * 48 + 1] = dk[1][r];
                vbase[r * VDIM * 3 + nt * 48 + 2] = dk[2][r];
            }
        }
    }
}

extern "C" void kernel_launch(void* const* d_in, const int* in_sizes, int n_in,
                              void* d_out, int out_size, void* d_ws, size_t ws_size,
                              hipStream_t stream) {
    const float* pos0 = (const float*)d_in[0];
    const float* pos1 = (const float*)d_in[1];
    const int*   src  = (const int*)d_in[2];
    const int*   dst  = (const int*)d_in[3];
    const float* Wa   = (const float*)d_in[4];
    const float* ba   = (const float*)d_in[5];
    const float* Wv   = (const float*)d_in[6];

    const int E = in_sizes[2];
    // out_size = B*E*(ADIM + VDIM*3)  =>  recover B; then N from pos_0 size.
    const int B = (int)((size_t)out_size / ((size_t)E * (ADIM + VDIM * 3)));
    const int N = in_sizes[0] / (B * 3);

    float* aout = (float*)d_out;
    float* vout = aout + (size_t)B * E * ADIM;

    const long long totalTiles = (long long)B * ((E + 15) / 16);
    const int wavesPerBlock = 256 / 32;
    long long blocksNeeded = (totalTiles + wavesPerBlock - 1) / wavesPerBlock;
    int blocks = (int)(blocksNeeded < 2048 ? blocksNeeded : 2048);
    if (blocks < 1) blocks = 1;

    if ((E & 15) == 0) {
        EdgeRelativeEmbed_kernel<false><<<blocks, 256, 0, stream>>>(
            pos0, pos1, src, dst, Wa, ba, Wv, aout, vout, B, N, E);
    } else {
        EdgeRelativeEmbed_kernel<true><<<blocks, 256, 0, stream>>>(
            pos0, pos1, src, dst, Wa, ba, Wv, aout, vout, B, N, E);
    }
}